// GAT_5978594476373
// MI455X (gfx1250) — compile-verified
//
#include <hip/hip_runtime.h>
#include <hip/hip_bf16.h>
#include <math.h>

// ---------------------------------------------------------------------------
// Types for CDNA5 WMMA f32 16x16x4: A = 16x4 (2 f32/lane), B = 4x16 (2 f32/lane),
// C/D = 16x16 f32 (8 f32/lane). wave32.
// ---------------------------------------------------------------------------
typedef float v2f __attribute__((ext_vector_type(2)));
typedef float v8f __attribute__((ext_vector_type(8)));

#define NEG_SLOPE 0.2f

// ---------------------------------------------------------------------------
// helpers
// ---------------------------------------------------------------------------
__device__ __forceinline__ float leaky(float v) {
    return v > 0.0f ? v : NEG_SLOPE * v;
}

// order-preserving monotone encoding of float into uint (for atomicMax).
// encode(v) >= 0x00800000 for any finite v, so sentinel 0 == "no edge".
__device__ __forceinline__ unsigned enc_f32(float v) {
    unsigned u = __float_as_uint(v);
    return (u >> 31) ? ~u : (u | 0x80000000u);
}
__device__ __forceinline__ float dec_max(unsigned u) {
    if (u == 0u) return 0.0f;  // node with no incoming edges -> reference uses 0
    unsigned b = (u >> 31) ? (u & 0x7fffffffu) : ~u;
    return __uint_as_float(b);
}

// ---------------------------------------------------------------------------
// init: zero accumulators, sentinel max buffers, seed d_out with bias b2
// ---------------------------------------------------------------------------
__global__ void gat_init(float* acc1, unsigned* emax1, float* denom1,
                         unsigned* emax2, float* denom2, float* out,
                         const float* __restrict__ b2, int N) {
    int idx = blockIdx.x * blockDim.x + threadIdx.x;
    if (idx >= N * 64) return;
    acc1[idx] = 0.0f;
    out[idx]  = b2[idx & 63];
    if (idx < N * 8) { emax1[idx] = 0u; denom1[idx] = 0.0f; }
    if (idx < N)     { emax2[idx] = 0u; denom2[idx] = 0.0f; }
}

// ---------------------------------------------------------------------------
// f32 WMMA GEMM: C[M,Nn] = A[M,K] @ B[K,Nn].  One wave per 16x16 tile,
// 4 waves per block.  M % 16 == 0, Nn % 16 == 0, K % 4 == 0.
// A layout per ISA 7.12.2 (16x4): lanes 0-15 hold K={k0,k0+1}, lanes 16-31
// hold K={k0+2,k0+3}, M = lane%16.  B (4x16): row striped across lanes.
// ---------------------------------------------------------------------------
__global__ __launch_bounds__(128)
void wmma_gemm_f32(const float* __restrict__ A, const float* __restrict__ B,
                   float* __restrict__ C, int M, int Nn, int K) {
    const int wave = threadIdx.x >> 5;
    const int lane = threadIdx.x & 31;
    const int tilesN = Nn >> 4;
    const int tile = blockIdx.x * 4 + wave;
    const int totalTiles = (M >> 4) * tilesN;
    if (tile >= totalTiles) return;

    const int m0 = (tile / tilesN) << 4;
    const int n0 = (tile % tilesN) << 4;
    const int half = lane >> 4;   // 0: K pair {0,1}, 1: K pair {2,3}
    const int l    = lane & 15;

    v8f c = {};
    for (int k0 = 0; k0 < K; k0 += 4) {
        const int ka = k0 + 2 * half;
        v2f a, b;
        a.x = A[(size_t)(m0 + l) * K + ka];
        a.y = A[(size_t)(m0 + l) * K + ka + 1];
        b.x = B[(size_t)ka * Nn + n0 + l];
        b.y = B[(size_t)(ka + 1) * Nn + n0 + l];
        c = __builtin_amdgcn_wmma_f32_16x16x4_f32(
                /*neg_a=*/false, a, /*neg_b=*/false, b,
                /*c_mod=*/(short)0, c, /*reuse_a=*/false, /*reuse_b=*/false);
    }

    const int row = m0 + half * 8;
#pragma unroll
    for (int i = 0; i < 8; ++i)
        C[(size_t)(row + i) * Nn + n0 + l] = c[i];
}

// ---------------------------------------------------------------------------
// layer-1 attention coefficients: el/er [N,8], feat [N,8,8]
// ---------------------------------------------------------------------------
__global__ void attn_coef1(const float* __restrict__ feat,
                           const float* __restrict__ al, const float* __restrict__ ar,
                           float* __restrict__ el, float* __restrict__ er, int N) {
    int idx = blockIdx.x * blockDim.x + threadIdx.x;  // over N*8
    if (idx >= N * 8) return;
    int n = idx >> 3, h = idx & 7;
    const float* f = feat + (size_t)n * 64 + h * 8;
    float sl = 0.0f, sr = 0.0f;
#pragma unroll
    for (int d = 0; d < 8; ++d) {
        sl += f[d] * al[h * 8 + d];
        sr += f[d] * ar[h * 8 + d];
    }
    el[idx] = sl;
    er[idx] = sr;
}

// layer-2: H=1, D=64
__global__ void attn_coef2(const float* __restrict__ feat,
                           const float* __restrict__ al, const float* __restrict__ ar,
                           float* __restrict__ el, float* __restrict__ er, int N) {
    int n = blockIdx.x * blockDim.x + threadIdx.x;
    if (n >= N) return;
    const float* f = feat + (size_t)n * 64;
    float sl = 0.0f, sr = 0.0f;
#pragma unroll
    for (int d = 0; d < 64; ++d) {
        sl += f[d] * al[d];
        sr += f[d] * ar[d];
    }
    el[n] = sl;
    er[n] = sr;
}

// ---------------------------------------------------------------------------
// edge pass 1: segment max of leaky(el[src]+er[dst]) over dst, per head
// ---------------------------------------------------------------------------
__global__ void edge_max(const int* __restrict__ src, const int* __restrict__ dst,
                         const float* __restrict__ el, const float* __restrict__ er,
                         unsigned* emax, int E, int H) {
    int idx = blockIdx.x * blockDim.x + threadIdx.x;  // over E*H
    if (idx >= E * H) return;
    int e = idx / H, h = idx - e * H;
    float v = leaky(el[src[e] * H + h] + er[dst[e] * H + h]);
    atomicMax(&emax[dst[e] * H + h], enc_f32(v));
}

// edge pass 2: denom[dst,h] += exp(v - max)
__global__ void edge_denom(const int* __restrict__ src, const int* __restrict__ dst,
                           const float* __restrict__ el, const float* __restrict__ er,
                           const unsigned* __restrict__ emax, float* denom, int E, int H) {
    int idx = blockIdx.x * blockDim.x + threadIdx.x;
    if (idx >= E * H) return;
    int e = idx / H, h = idx - e * H;
    int dn = dst[e];
    float v = leaky(el[src[e] * H + h] + er[dn * H + h]);
    float m = dec_max(emax[dn * H + h]);
    atomicAdd(&denom[dn * H + h], expf(v - m));
}

// edge pass 3, layer 1: acc1[dst,h,0:8] += feat1[src,h,0:8] * alpha
__global__ void edge_agg1(const int* __restrict__ src, const int* __restrict__ dst,
                          const float* __restrict__ el, const float* __restrict__ er,
                          const unsigned* __restrict__ emax, const float* __restrict__ denom,
                          const float* __restrict__ feat, float* acc, int E) {
    int idx = blockIdx.x * blockDim.x + threadIdx.x;  // over E*8
    if (idx >= E * 8) return;
    int e = idx >> 3, h = idx & 7;
    int s = src[e], dn = dst[e];
    float v = leaky(el[s * 8 + h] + er[dn * 8 + h]);
    float m = dec_max(emax[dn * 8 + h]);
    float alpha = expf(v - m) / fmaxf(denom[dn * 8 + h], 1e-9f);
    const float* fs = feat + (size_t)s * 64 + h * 8;
    float* od = acc + (size_t)dn * 64 + h * 8;
#pragma unroll
    for (int d = 0; d < 8; ++d)
        atomicAdd(&od[d], fs[d] * alpha);
}

// edge pass 3, layer 2 (H=1, 64 feats): 4 chunks of 16 per edge
__global__ void edge_agg2(const int* __restrict__ src, const int* __restrict__ dst,
                          const float* __restrict__ el, const float* __restrict__ er,
                          const unsigned* __restrict__ emax, const float* __restrict__ denom,
                          const float* __restrict__ feat, float* out, int E) {
    int idx = blockIdx.x * blockDim.x + threadIdx.x;  // over E*4
    if (idx >= E * 4) return;
    int e = idx >> 2, chunk = idx & 3;
    int s = src[e], dn = dst[e];
    float v = leaky(el[s] + er[dn]);
    float m = dec_max(emax[dn]);
    float alpha = expf(v - m) / fmaxf(denom[dn], 1e-9f);
    const float* fs = feat + (size_t)s * 64 + chunk * 16;
    float* od = out + (size_t)dn * 64 + chunk * 16;
#pragma unroll
    for (int d = 0; d < 16; ++d)
        atomicAdd(&od[d], fs[d] * alpha);
}

// ---------------------------------------------------------------------------
// between layers: x2 = elu(acc1 + b1)
// ---------------------------------------------------------------------------
__global__ void elu_bias(const float* __restrict__ acc, const float* __restrict__ b1,
                         float* __restrict__ x2, int N) {
    int idx = blockIdx.x * blockDim.x + threadIdx.x;  // over N*64
    if (idx >= N * 64) return;
    float t = acc[idx] + b1[idx & 63];
    x2[idx] = t > 0.0f ? t : expm1f(t);
}

// ---------------------------------------------------------------------------
// launcher
// ---------------------------------------------------------------------------
extern "C" void kernel_launch(void* const* d_in, const int* in_sizes, int n_in,
                              void* d_out, int out_size, void* d_ws, size_t ws_size,
                              hipStream_t stream) {
    (void)n_in; (void)out_size; (void)ws_size;
    const float* h   = (const float*)d_in[0];
    const int*   src = (const int*)d_in[1];
    const int*   dst = (const int*)d_in[2];
    const float* W1  = (const float*)d_in[3];
    const float* al1 = (const float*)d_in[4];
    const float* ar1 = (const float*)d_in[5];
    const float* b1  = (const float*)d_in[6];
    const float* W2  = (const float*)d_in[7];
    const float* al2 = (const float*)d_in[8];
    const float* ar2 = (const float*)d_in[9];
    const float* b2  = (const float*)d_in[10];
    float* out = (float*)d_out;

    const int N = in_sizes[0] / 128;   // 50000
    const int E = in_sizes[1];         // 800000

    // workspace layout
    float*    feat1  = (float*)d_ws;                        // N*64
    float*    el1    = feat1 + (size_t)N * 64;              // N*8
    float*    er1    = el1 + (size_t)N * 8;                 // N*8
    unsigned* emax1  = (unsigned*)(er1 + (size_t)N * 8);    // N*8
    float*    denom1 = (float*)(emax1 + (size_t)N * 8);     // N*8
    float*    acc1   = denom1 + (size_t)N * 8;              // N*64
    float*    x2     = acc1 + (size_t)N * 64;               // N*64
    float*    feat2  = x2 + (size_t)N * 64;                 // N*64
    float*    el2    = feat2 + (size_t)N * 64;              // N
    float*    er2    = el2 + N;                             // N
    unsigned* emax2  = (unsigned*)(er2 + N);                // N
    float*    denom2 = (float*)(emax2 + N);                 // N

    const int B = 256;
    const int gN64 = (N * 64 + B - 1) / B;
    const int gN8  = (N * 8 + B - 1) / B;
    const int gN   = (N + B - 1) / B;
    const int gE8  = (E * 8 + B - 1) / B;
    const int gE4  = (E * 4 + B - 1) / B;
    const int gE1  = (E + B - 1) / B;

    // 0) init accumulators / sentinels / output=b2
    gat_init<<<gN64, B, 0, stream>>>(acc1, emax1, denom1, emax2, denom2, out, b2, N);

    // ---- layer 1 ----
    {
        int tiles = (N / 16) * (64 / 16);
        wmma_gemm_f32<<<(tiles + 3) / 4, 128, 0, stream>>>(h, W1, feat1, N, 64, 128);
    }
    attn_coef1<<<gN8, B, 0, stream>>>(feat1, al1, ar1, el1, er1, N);
    edge_max  <<<gE8, B, 0, stream>>>(src, dst, el1, er1, emax1, E, 8);
    edge_denom<<<gE8, B, 0, stream>>>(src, dst, el1, er1, emax1, denom1, E, 8);
    edge_agg1 <<<gE8, B, 0, stream>>>(src, dst, el1, er1, emax1, denom1, feat1, acc1, E);
    elu_bias  <<<gN64, B, 0, stream>>>(acc1, b1, x2, N);

    // ---- layer 2 ----
    {
        int tiles = (N / 16) * (64 / 16);
        wmma_gemm_f32<<<(tiles + 3) / 4, 128, 0, stream>>>(x2, W2, feat2, N, 64, 64);
    }
    attn_coef2<<<gN, B, 0, stream>>>(feat2, al2, ar2, el2, er2, N);
    edge_max  <<<gE1, B, 0, stream>>>(src, dst, el2, er2, emax2, E, 1);
    edge_denom<<<gE1, B, 0, stream>>>(src, dst, el2, er2, emax2, denom2, E, 1);
    edge_agg2 <<<gE4, B, 0, stream>>>(src, dst, el2, er2, emax2, denom2, feat2, out, E);
}